// detector_33105607918057
// MI455X (gfx1250) — compile-verified
//
#include <hip/hip_runtime.h>
#include <stdint.h>

#define NB   10
#define NN   512
#define NC   37
#define ND   2048
#define MAXDET 100
#define ROWLEN 2053        // 4 box + 1 score + 2048 feats
#define NCLS 36            // classes 1..36
#define SCORE_TH 0.1f
#define NMS_TH   0.4f

// ---------------------------------------------------------------------------
// Kernel 1: per (b, class) block — sort, decode, NMS, compact.
// Writes the 5-float header of every one of the 100 slots and a slot->roi
// index table (-1 = empty) used by the gather kernel.
// ---------------------------------------------------------------------------
__global__ __launch_bounds__(256) void nms_kernel(
    const float* __restrict__ boxes,      // (B,N,4)
    const float* __restrict__ cls_prob,   // (B,N,C)
    const float* __restrict__ bbox_pred,  // (B,N,4C)
    const float* __restrict__ im_info,    // (B,3)
    float* __restrict__ out,              // (B,36,100,2053)
    int*   __restrict__ slot_idx)         // B*36*100
{
    __shared__ float  s_sc[NN];
    __shared__ int    s_or[NN];
    __shared__ float4 s_bx[NN];
    __shared__ float  s_area[NN];
    __shared__ int    s_keep[NN];
    __shared__ int    s_rank[NN];
    __shared__ int    s_count;

    const int blk = blockIdx.x;
    const int b   = blk / NCLS;
    const int cls = (blk % NCLS) + 1;      // class ids 1..36
    const int tid = threadIdx.x;

    // --- load scores for this class -------------------------------------
    for (int e = 0; e < 2; ++e) {
        int n = tid + (e << 8);
        s_sc[n] = cls_prob[((size_t)b * NN + n) * NC + cls];
        s_or[n] = n;
    }

    // --- bitonic sort: descending score, ties -> ascending original idx --
    // (matches jax stable argsort(-scores))
    for (int k = 2; k <= NN; k <<= 1) {
        for (int j = k >> 1; j > 0; j >>= 1) {
            __syncthreads();
            for (int e = 0; e < 2; ++e) {
                int i   = tid + (e << 8);
                int ixj = i ^ j;
                if (ixj > i) {
                    float si = s_sc[i], sj = s_sc[ixj];
                    int   oi = s_or[i], oj = s_or[ixj];
                    // "i belongs after ixj" in the desired (descending) order
                    bool i_after_j = (si < sj) || (si == sj && oi > oj);
                    if (((i & k) == 0) == i_after_j) {
                        s_sc[i] = sj; s_sc[ixj] = si;
                        s_or[i] = oj; s_or[ixj] = oi;
                    }
                }
            }
        }
    }
    __syncthreads();

    // --- decode + clip boxes at sorted positions -------------------------
    const float hmax      = im_info[b * 3 + 0] - 1.0f;
    const float wmax      = im_info[b * 3 + 1] - 1.0f;
    const float inv_scale = 1.0f / im_info[2];     // reference uses im_info[0,2]

    for (int e = 0; e < 2; ++e) {
        int p = tid + (e << 8);
        int n = s_or[p];
        const float* bb = boxes + ((size_t)b * NN + n) * 4;
        float x1 = bb[0], y1 = bb[1], x2 = bb[2], y2 = bb[3];
        float w  = x2 - x1 + 1.0f, h = y2 - y1 + 1.0f;
        float cx = x1 + 0.5f * w,  cy = y1 + 0.5f * h;
        const float* dp = bbox_pred + ((size_t)b * NN + n) * (4 * NC) + 4 * cls;
        float dx = dp[0] * 0.1f, dy = dp[1] * 0.1f;
        float dw = dp[2] * 0.2f, dh = dp[3] * 0.2f;
        float pcx = dx * w + cx, pcy = dy * h + cy;
        float pw  = expf(dw) * w, ph = expf(dh) * h;
        float nx1 = fminf(fmaxf(pcx - 0.5f * pw, 0.f), wmax) * inv_scale;
        float ny1 = fminf(fmaxf(pcy - 0.5f * ph, 0.f), hmax) * inv_scale;
        float nx2 = fminf(fmaxf(pcx + 0.5f * pw, 0.f), wmax) * inv_scale;
        float ny2 = fminf(fmaxf(pcy + 0.5f * ph, 0.f), hmax) * inv_scale;
        s_bx[p]   = make_float4(nx1, ny1, nx2, ny2);
        s_area[p] = (nx2 - nx1) * (ny2 - ny1);
        s_keep[p] = (s_sc[p] > SCORE_TH) ? 1 : 0;
    }

    // --- greedy NMS: sequential over i, parallel over j>i ----------------
    for (int i = 0; i < NN; ++i) {
        __syncthreads();                 // make prior suppressions visible
        if (!s_keep[i]) continue;        // uniform: same decision in all lanes
        float4 bi = s_bx[i];
        float  ai = s_area[i];
        for (int e = 0; e < 2; ++e) {
            int p = tid + (e << 8);
            if (p > i && s_keep[p]) {
                float4 bp = s_bx[p];
                float iw = fmaxf(fminf(bi.z, bp.z) - fmaxf(bi.x, bp.x), 0.f);
                float ih = fmaxf(fminf(bi.w, bp.w) - fmaxf(bi.y, bp.y), 0.f);
                float inter = iw * ih;
                float iou   = inter / (ai + s_area[p] - inter + 1e-9f);
                if (iou > NMS_TH) s_keep[p] = 0;
            }
        }
    }
    __syncthreads();

    // --- serial rank scan (cheap: 512 iters) -----------------------------
    // handles both the top-100 cap and the class==1 single-keep rule
    if (tid == 0) {
        int r = 0;
        for (int p = 0; p < NN; ++p) {
            if (s_keep[p]) {
                bool take = (r < MAXDET) && !(cls == 1 && r >= 1);
                if (take) s_rank[p] = r++;
                else      s_keep[p] = 0;
            }
        }
        s_count = r;
    }
    __syncthreads();

    // --- emit 5-float headers + slot index table (streaming stores) ------
    const size_t row_base = ((size_t)b * NCLS + (cls - 1)) * MAXDET;
    for (int e = 0; e < 2; ++e) {
        int p = tid + (e << 8);
        if (s_keep[p]) {
            int slot = s_rank[p];
            float* o = out + (row_base + slot) * (size_t)ROWLEN;
            float4 bx = s_bx[p];
            __builtin_nontemporal_store(fmaxf(bx.x, 0.f), o + 0);
            __builtin_nontemporal_store(fmaxf(bx.y, 0.f), o + 1);
            __builtin_nontemporal_store(fmaxf(bx.z, 0.f), o + 2);
            __builtin_nontemporal_store(fmaxf(bx.w, 0.f), o + 3);
            __builtin_nontemporal_store(s_sc[p],          o + 4);
            slot_idx[row_base + slot] = s_or[p];
        }
    }
    int cnt = s_count;
    if (tid < MAXDET && tid >= cnt) {
        float* o = out + (row_base + tid) * (size_t)ROWLEN;
        __builtin_nontemporal_store(0.f, o + 0);
        __builtin_nontemporal_store(0.f, o + 1);
        __builtin_nontemporal_store(0.f, o + 2);
        __builtin_nontemporal_store(0.f, o + 3);
        __builtin_nontemporal_store(0.f, o + 4);
        slot_idx[row_base + tid] = -1;
    }
}

// ---------------------------------------------------------------------------
// Kernel 2: feature gather — one block per output row.
// CDNA5 async data mover (GLOBAL_LOAD_ASYNC_TO_LDS_B128, ASYNCcnt-tracked)
// stages the 8 KB feature row in LDS; features are then written out with
// coalesced non-temporal b32 stores (the 2053-float row stride leaves the
// feature section only 4B-aligned, and the 295 MB output must not thrash the
// 192 MB L2 that holds the reusable 42 MB roi_features).
// ---------------------------------------------------------------------------
__global__ __launch_bounds__(256) void gather_kernel(
    const float* __restrict__ feats,      // (B,N,2048)
    const int*   __restrict__ slot_idx,   // B*36*100
    float*       __restrict__ out)
{
    __shared__ float sfeat[ND];
    const int row = blockIdx.x;                 // 0 .. 35999
    const int b   = row / (NCLS * MAXDET);
    const int tid = threadIdx.x;
    const int idx = slot_idx[row];
    float* dst = out + (size_t)row * ROWLEN + 5;

    if (idx < 0) {                              // empty slot: zero-fill
        #pragma unroll
        for (int k = 0; k < 8; ++k)
            __builtin_nontemporal_store(0.f, dst + tid + (k << 8));
        return;
    }

    const float* src = feats + ((size_t)b * NN + idx) * ND;   // 16B aligned
    unsigned voff = tid * 32u;                                // 32 B / thread
    unsigned ldsa = (unsigned)(uintptr_t)(&sfeat[tid * 8]);   // LDS byte addr
    asm volatile(
        "global_load_async_to_lds_b128 %0, %1, %2\n\t"
        "global_load_async_to_lds_b128 %0, %1, %2 offset:16\n\t"
        "s_wait_asynccnt 0x0"
        :: "v"(ldsa), "v"(voff), "s"(src)
        : "memory");
    __syncthreads();

    #pragma unroll
    for (int k = 0; k < 8; ++k)
        __builtin_nontemporal_store(sfeat[tid + (k << 8)], dst + tid + (k << 8));
}

// ---------------------------------------------------------------------------
extern "C" void kernel_launch(void* const* d_in, const int* in_sizes, int n_in,
                              void* d_out, int out_size, void* d_ws, size_t ws_size,
                              hipStream_t stream) {
    const float* boxes     = (const float*)d_in[0];
    const float* cls_prob  = (const float*)d_in[1];
    const float* bbox_pred = (const float*)d_in[2];
    const float* feats     = (const float*)d_in[3];
    const float* im_info   = (const float*)d_in[4];
    float* out    = (float*)d_out;
    int*   sl_idx = (int*)d_ws;                 // 36000 ints = 144 KB

    nms_kernel<<<NB * NCLS, 256, 0, stream>>>(boxes, cls_prob, bbox_pred,
                                              im_info, out, sl_idx);
    gather_kernel<<<NB * NCLS * MAXDET, 256, 0, stream>>>(feats, sl_idx, out);
}